// SelfAttention_24292335026749
// MI455X (gfx1250) — compile-verified
//
#include <hip/hip_runtime.h>

typedef float v2f __attribute__((ext_vector_type(2)));
typedef float v8f __attribute__((ext_vector_type(8)));

#define N_  32
#define V_  2000
#define C_  256
#define H_  128
#define CHUNKS 25
#define ROWS_PER_CHUNK 80                 // 25 * 80 = 2000
#define SCALE 0.08838834764831845f        // 128^-0.5

// ---------------------------------------------------------------------------
// Pass 1a: partial[chunk][n][c] = sum_{v in chunk} to_attn[v] * x[n,v,c]
// 64 threads (2 waves), each thread owns 4 channels -> pure b128 streaming.
// ---------------------------------------------------------------------------
__global__ void k_partial_y(const float* __restrict__ x,
                            const float* __restrict__ to_attn,
                            float* __restrict__ partial) {
  const int chunk = blockIdx.x;
  const int n     = blockIdx.y;
  const int c4    = threadIdx.x * 4;               // 64 threads * 4 = 256 = C_
  const int v0    = chunk * ROWS_PER_CHUNK;
  const float* xp = x + ((size_t)n * V_ + v0) * C_ + c4;
  float4 acc = make_float4(0.f, 0.f, 0.f, 0.f);
  for (int vi = 0; vi < ROWS_PER_CHUNK; ++vi) {
    const float  t  = to_attn[v0 + vi];
    const float4 xv = *reinterpret_cast<const float4*>(xp);
    __builtin_prefetch((const void*)(xp + 4 * C_), 0, 1);
    acc.x += t * xv.x; acc.y += t * xv.y;
    acc.z += t * xv.z; acc.w += t * xv.w;
    xp += C_;
  }
  float* pp = partial + ((size_t)(chunk * N_ + n)) * C_ + c4;
  *reinterpret_cast<float4*>(pp) = acc;
}

// ---------------------------------------------------------------------------
// Pass 1b: y[n,c] = sum over 25 chunks (fixed order -> deterministic)
// ---------------------------------------------------------------------------
__global__ void k_reduce_y(const float* __restrict__ partial,
                           float* __restrict__ y) {
  const int i = blockIdx.x * blockDim.x + threadIdx.x;   // N_*C_ = 8192
  float acc = 0.f;
  for (int j = 0; j < CHUNKS; ++j) acc += partial[(size_t)j * N_ * C_ + i];
  y[i] = acc;
}

// ---------------------------------------------------------------------------
// Pass 2a (WMMA f32): s[32,128] = y[32,256] @ to_k[256,128]
// One wave per 16x16 output tile; K-loop of V_WMMA_F32_16X16X4_F32.
// A layout (16x4 f32): lanes 0-15 -> M=0..15; VGPR0 = K(0|2), VGPR1 = K(1|3)
// split by lane-half. B layout mirrors it per K-row.
// ---------------------------------------------------------------------------
__global__ void k_gemm1(const float* __restrict__ y,
                        const float* __restrict__ to_k,
                        float* __restrict__ s) {
  const int lane = threadIdx.x;          // 32 threads, EXEC all ones
  const int m0   = (blockIdx.x & 1) * 16;        // 2 M tiles
  const int h0   = (blockIdx.x >> 1) * 16;       // 8 N tiles
  const int r    = lane & 15;
  const int g    = lane >> 4;
  const int row  = m0 + r;
  v8f acc = {0.f, 0.f, 0.f, 0.f, 0.f, 0.f, 0.f, 0.f};
  for (int k0 = 0; k0 < C_; k0 += 4) {
    const int ka = k0 + 2 * g;
    v2f a, b;
    a.x = y[row * C_ + ka];
    a.y = y[row * C_ + ka + 1];
    b.x = to_k[(ka)     * H_ + h0 + r];
    b.y = to_k[(ka + 1) * H_ + h0 + r];
    acc = __builtin_amdgcn_wmma_f32_16x16x4_f32(
        /*neg_a=*/false, a, /*neg_b=*/false, b,
        /*c_mod=*/(short)0, acc, /*reuse_a=*/false, /*reuse_b=*/false);
  }
  for (int rr = 0; rr < 8; ++rr)
    s[(m0 + rr + 8 * g) * H_ + h0 + r] = acc[rr];
}

// ---------------------------------------------------------------------------
// Pass 2b (WMMA f32): w[32,256] = s[32,128] @ to_q^T, scale folded in.
// B[k][c] = to_q[c*H + k].
// ---------------------------------------------------------------------------
__global__ void k_gemm2(const float* __restrict__ s,
                        const float* __restrict__ to_q,
                        float* __restrict__ w) {
  const int lane = threadIdx.x;
  const int m0   = (blockIdx.x & 1) * 16;        // 2 M tiles
  const int c0   = (blockIdx.x >> 1) * 16;       // 16 N tiles
  const int r    = lane & 15;
  const int g    = lane >> 4;
  const int row  = m0 + r;
  v8f acc = {0.f, 0.f, 0.f, 0.f, 0.f, 0.f, 0.f, 0.f};
  for (int k0 = 0; k0 < H_; k0 += 4) {
    const int ka = k0 + 2 * g;
    v2f a, b;
    a.x = s[row * H_ + ka];
    a.y = s[row * H_ + ka + 1];
    b.x = to_q[(c0 + r) * H_ + ka];
    b.y = to_q[(c0 + r) * H_ + ka + 1];
    acc = __builtin_amdgcn_wmma_f32_16x16x4_f32(
        false, a, false, b, (short)0, acc, false, false);
  }
  for (int rr = 0; rr < 8; ++rr)
    w[(m0 + rr + 8 * g) * C_ + c0 + r] = acc[rr] * SCALE;
}

// ---------------------------------------------------------------------------
// Pass 3: attn_raw[n,v] = x[n,v,:] . w[n,:]   (w pre-scaled)
// 8 waves/block, one wave32 per row; lane holds 8 channels of w in registers.
// ---------------------------------------------------------------------------
__global__ void k_dot(const float* __restrict__ x,
                      const float* __restrict__ w,
                      float* __restrict__ attn_raw) {
  const int chunk = blockIdx.x;
  const int n     = blockIdx.y;
  const int lane  = threadIdx.x & 31;
  const int wave  = threadIdx.x >> 5;            // 0..7
  const float4 w0 = *reinterpret_cast<const float4*>(w + n * C_ + lane * 8);
  const float4 w1 = *reinterpret_cast<const float4*>(w + n * C_ + lane * 8 + 4);
  for (int vi = wave; vi < ROWS_PER_CHUNK; vi += 8) {
    const int v = chunk * ROWS_PER_CHUNK + vi;
    const float* xp = x + ((size_t)n * V_ + v) * C_ + lane * 8;
    const float4 a = *reinterpret_cast<const float4*>(xp);
    const float4 b = *reinterpret_cast<const float4*>(xp + 4);
    float sum = a.x * w0.x + a.y * w0.y + a.z * w0.z + a.w * w0.w
              + b.x * w1.x + b.y * w1.y + b.z * w1.z + b.w * w1.w;
    for (int off = 16; off; off >>= 1) sum += __shfl_xor(sum, off, 32);
    if (lane == 0) attn_raw[n * V_ + v] = sum;
  }
}

// ---------------------------------------------------------------------------
// Pass 4: L1 normalize over V per batch (deterministic tree reduce).
// ---------------------------------------------------------------------------
__global__ void k_normdiv(const float* __restrict__ attn_raw,
                          float* __restrict__ out) {
  __shared__ float red[256];
  const int n   = blockIdx.x;
  const int tid = threadIdx.x;
  float acc = 0.f;
  for (int i = tid; i < V_; i += 256) acc += fabsf(attn_raw[n * V_ + i]);
  red[tid] = acc;
  __syncthreads();
  for (int sft = 128; sft; sft >>= 1) {
    if (tid < sft) red[tid] += red[tid + sft];
    __syncthreads();
  }
  const float inv = 1.0f / fmaxf(red[0], 1e-12f);
  for (int i = tid; i < V_; i += 256)
    out[n * V_ + i] = attn_raw[n * V_ + i] * inv;
}

// ---------------------------------------------------------------------------
extern "C" void kernel_launch(void* const* d_in, const int* in_sizes, int n_in,
                              void* d_out, int out_size, void* d_ws, size_t ws_size,
                              hipStream_t stream) {
  const float* x       = (const float*)d_in[0];   // [32,2000,256]
  const float* to_q    = (const float*)d_in[1];   // [256,128]
  const float* to_k    = (const float*)d_in[2];   // [256,128]
  const float* to_attn = (const float*)d_in[3];   // [2000]
  float* out = (float*)d_out;                     // [32,2000]

  float* ws       = (float*)d_ws;
  float* partial  = ws;                           // 25*32*256
  float* y        = partial + CHUNKS * N_ * C_;   // 32*256
  float* s        = y + N_ * C_;                  // 32*128
  float* w        = s + N_ * H_;                  // 32*256
  float* attn_raw = w + N_ * C_;                  // 32*2000

  k_partial_y<<<dim3(CHUNKS, N_), 64, 0, stream>>>(x, to_attn, partial);
  k_reduce_y<<<(N_ * C_) / 256, 256, 0, stream>>>(partial, y);
  k_gemm1<<<16, 32, 0, stream>>>(y, to_k, s);
  k_gemm2<<<32, 32, 0, stream>>>(s, to_q, w);
  k_dot<<<dim3(CHUNKS, N_), 256, 0, stream>>>(x, w, attn_raw);
  k_normdiv<<<N_, 256, 0, stream>>>(attn_raw, out);
}